// FG_MSA_31525059953275
// MI455X (gfx1250) — compile-verified
//
#include <hip/hip_runtime.h>
#include <math.h>

// ---------------------------------------------------------------------------
// FG_MSA for MI455X (gfx1250): all GEMMs on V_WMMA_F32_16X16X4_F32 (fp32 exact)
// ---------------------------------------------------------------------------

typedef __attribute__((ext_vector_type(2))) float v2f;
typedef __attribute__((ext_vector_type(8))) float v8f;

#define WMMA_F32(a, b, c) \
  __builtin_amdgcn_wmma_f32_16x16x4_f32(false, (a), false, (b), (short)0, (c), false, false)

// ======================= generic 512-K / 512-N GEMM ========================
// C[row, 0:512] = A[row, 0:512] @ W[0:512, 0:512]   (optionally batched W)
// MODE 0: C = acc
// MODE 1: C = (acc + R[row,col]) * Mul[row,col]        (prior residual * v_inp)
// MODE 2: C = acc + bias[col] + C[row,col]             (proj bias + conv branch)
template <int MODE>
__global__ __launch_bounds__(256) void gemm512_wmma(
    const float* __restrict__ A, const float* __restrict__ W, float* __restrict__ C,
    int m_per_z, long w_zstride, const float* __restrict__ R,
    const float* __restrict__ Mul, const float* __restrict__ bias) {
  __shared__ float As[2][128 * 17];   // [m][k], stride 17 kills bank conflicts
  __shared__ float Ws[2][16 * 144];   // [k][n], stride 144 (+32 banks per 2 rows)

  const int tid = threadIdx.x, lane = tid & 31, wave = tid >> 5;
  const int wm = wave >> 1, wn = wave & 1;            // waves 4 (M) x 2 (N)
  const int nbase = blockIdx.y * 128;
  const long rowz0 = (long)blockIdx.z * m_per_z + (long)blockIdx.x * 128;
  const float* Az = A + rowz0 * 512;
  const float* Wz = W + (long)blockIdx.z * w_zstride + nbase;

  const v8f z8 = {0.f, 0.f, 0.f, 0.f, 0.f, 0.f, 0.f, 0.f};
  v8f acc[2][4];
#pragma unroll
  for (int i = 0; i < 2; i++)
#pragma unroll
    for (int j = 0; j < 4; j++) acc[i][j] = z8;

  auto load_tiles = [&](int kb, int buf) {
#pragma unroll
    for (int i = 0; i < 2; i++) {
      int f = tid + i * 256;
      int r = f >> 2, kc = (f & 3) << 2;
      float4 va = *(const float4*)(Az + (long)r * 512 + kb * 16 + kc);
      float* pa = &As[buf][r * 17 + kc];
      pa[0] = va.x; pa[1] = va.y; pa[2] = va.z; pa[3] = va.w;
      int k = f >> 5, n4 = (f & 31) << 2;
      float4 vw = *(const float4*)(Wz + (long)(kb * 16 + k) * 512 + n4);
      float* pw = &Ws[buf][k * 144 + n4];
      pw[0] = vw.x; pw[1] = vw.y; pw[2] = vw.z; pw[3] = vw.w;
    }
  };

  const int ml = lane & 15, kq = (lane >> 4) << 1;
  load_tiles(0, 0);
  __syncthreads();
  for (int kb = 0; kb < 32; kb++) {
    int cur = kb & 1;
    if (kb < 31) load_tiles(kb + 1, cur ^ 1);
#pragma unroll
    for (int kk = 0; kk < 16; kk += 4) {
      v2f af[2], bf[4];
#pragma unroll
      for (int i = 0; i < 2; i++) {
        int m = wm * 32 + i * 16 + ml;
        af[i].x = As[cur][m * 17 + kk + kq];
        af[i].y = As[cur][m * 17 + kk + kq + 1];
      }
#pragma unroll
      for (int j = 0; j < 4; j++) {
        int n = wn * 64 + j * 16 + ml;
        bf[j].x = Ws[cur][(kk + kq) * 144 + n];
        bf[j].y = Ws[cur][(kk + kq + 1) * 144 + n];
      }
#pragma unroll
      for (int i = 0; i < 2; i++)
#pragma unroll
        for (int j = 0; j < 4; j++) acc[i][j] = WMMA_F32(af[i], bf[j], acc[i][j]);
    }
    __syncthreads();
  }

  const int mh = (lane >> 4) << 3, nl = lane & 15;
#pragma unroll
  for (int i = 0; i < 2; i++)
#pragma unroll
    for (int j = 0; j < 4; j++)
#pragma unroll
      for (int r = 0; r < 8; r++) {
        long row = rowz0 + wm * 32 + i * 16 + mh + r;
        int colg = nbase + wn * 64 + j * 16 + nl;
        long o = row * 512 + colg;
        float v = acc[i][j][r];
        if (MODE == 1) v = (v + R[o]) * Mul[o];
        else if (MODE == 2) v = v + bias[colg] + C[o];
        C[o] = v;
      }
}

// ======================= LayerNorm over rows of 512 ========================
__global__ __launch_bounds__(256) void ln_rows_k(float* __restrict__ X,
                                                 const float* __restrict__ g,
                                                 const float* __restrict__ b) {
  int tid = threadIdx.x, lane = tid & 31, wave = tid >> 5;
  long row = (long)blockIdx.x * 8 + wave;
  float* xr = X + row * 512;
  float v[16], s = 0.f, s2 = 0.f;
#pragma unroll
  for (int i = 0; i < 16; i++) { v[i] = xr[lane + i * 32]; s += v[i]; s2 += v[i] * v[i]; }
#pragma unroll
  for (int o = 16; o >= 1; o >>= 1) { s += __shfl_xor(s, o, 32); s2 += __shfl_xor(s2, o, 32); }
  float mu = s * (1.0f / 512.0f);
  float var = s2 * (1.0f / 512.0f) - mu * mu;
  float rstd = rsqrtf(var + 1e-5f);
#pragma unroll
  for (int i = 0; i < 16; i++) { int c = lane + i * 32; xr[c] = (v[i] - mu) * rstd * g[c] + b[c]; }
}

// ========= SIF Gram: Ab[z,i,j] = (1/sqrt(512)) * sum_n Q[z,n,i]*K[z,n,j] =========
__global__ __launch_bounds__(256) void sif_gram_wmma(const float* __restrict__ Q,
                                                     const float* __restrict__ Kn,
                                                     float* __restrict__ Ab) {
  __shared__ float Qs[2][16 * 144];
  __shared__ float Ks[2][16 * 144];
  const int tid = threadIdx.x, lane = tid & 31, wave = tid >> 5;
  const int wm = wave >> 1, wn = wave & 1;
  const int z = blockIdx.z, ibase = blockIdx.x * 128, jbase = blockIdx.y * 128;
  const float* Qz = Q + (long)z * 4096 * 512 + ibase;
  const float* Kz = Kn + (long)z * 4096 * 512 + jbase;

  const v8f z8 = {0.f, 0.f, 0.f, 0.f, 0.f, 0.f, 0.f, 0.f};
  v8f acc[2][4];
#pragma unroll
  for (int i = 0; i < 2; i++)
#pragma unroll
    for (int j = 0; j < 4; j++) acc[i][j] = z8;

  auto load_tiles = [&](int kb, int buf) {
#pragma unroll
    for (int i = 0; i < 2; i++) {
      int f = tid + i * 256;
      int n = f >> 5, c4 = (f & 31) << 2;
      float4 vq = *(const float4*)(Qz + (long)(kb * 16 + n) * 512 + c4);
      float* pq = &Qs[buf][n * 144 + c4];
      pq[0] = vq.x; pq[1] = vq.y; pq[2] = vq.z; pq[3] = vq.w;
      float4 vk = *(const float4*)(Kz + (long)(kb * 16 + n) * 512 + c4);
      float* pk = &Ks[buf][n * 144 + c4];
      pk[0] = vk.x; pk[1] = vk.y; pk[2] = vk.z; pk[3] = vk.w;
    }
  };

  const int ml = lane & 15, kq = (lane >> 4) << 1;
  load_tiles(0, 0);
  __syncthreads();
  for (int kb = 0; kb < 256; kb++) {
    int cur = kb & 1;
    if (kb < 255) load_tiles(kb + 1, cur ^ 1);
#pragma unroll
    for (int kk = 0; kk < 16; kk += 4) {
      v2f af[2], bf[4];
#pragma unroll
      for (int i = 0; i < 2; i++) {               // A[m][k] = Q[n=k][c=m]
        int m = wm * 32 + i * 16 + ml;
        af[i].x = Qs[cur][(kk + kq) * 144 + m];
        af[i].y = Qs[cur][(kk + kq + 1) * 144 + m];
      }
#pragma unroll
      for (int j = 0; j < 4; j++) {
        int n = wn * 64 + j * 16 + ml;
        bf[j].x = Ks[cur][(kk + kq) * 144 + n];
        bf[j].y = Ks[cur][(kk + kq + 1) * 144 + n];
      }
#pragma unroll
      for (int i = 0; i < 2; i++)
#pragma unroll
        for (int j = 0; j < 4; j++) acc[i][j] = WMMA_F32(af[i], bf[j], acc[i][j]);
    }
    __syncthreads();
  }

  const int mh = (lane >> 4) << 3, nl = lane & 15;
  const float scale = 0.0441941738f;  // 1/sqrt(512)
#pragma unroll
  for (int i = 0; i < 2; i++)
#pragma unroll
    for (int j = 0; j < 4; j++)
#pragma unroll
      for (int r = 0; r < 8; r++) {
        int row = ibase + wm * 32 + i * 16 + mh + r;
        int col = jbase + wn * 64 + j * 16 + nl;
        Ab[(long)z * 262144 + (long)row * 512 + col] = acc[i][j][r] * scale;
      }
}

// ====== row softmax over Ab[z,i,:], stored transposed: SmT[z,j,i] ======
__global__ __launch_bounds__(256) void softmaxT_k(const float* __restrict__ Ab,
                                                  float* __restrict__ SmT) {
  __shared__ float T[16 * 513];
  int tid = threadIdx.x, lane = tid & 31, wave = tid >> 5;
  int z = blockIdx.y, rb = blockIdx.x;
  const float* base = Ab + (long)z * 262144 + (long)rb * 16 * 512;
  for (int s = 0; s < 2; s++) {
    int rl = wave * 2 + s;
    float v[16], m = -3.4e38f;
#pragma unroll
    for (int i = 0; i < 16; i++) { v[i] = base[rl * 512 + lane + i * 32]; m = fmaxf(m, v[i]); }
#pragma unroll
    for (int o = 16; o >= 1; o >>= 1) m = fmaxf(m, __shfl_xor(m, o, 32));
    float sum = 0.f;
#pragma unroll
    for (int i = 0; i < 16; i++) { v[i] = expf(v[i] - m); sum += v[i]; }
#pragma unroll
    for (int o = 16; o >= 1; o >>= 1) sum += __shfl_xor(sum, o, 32);
    float inv = 1.0f / sum;
#pragma unroll
    for (int i = 0; i < 16; i++) T[rl * 513 + lane + i * 32] = v[i] * inv;
  }
  __syncthreads();
  float* outb = SmT + (long)z * 262144 + rb * 16;
  int ii = tid & 15, jg = tid >> 4;
  for (int jj = 0; jj < 32; jj++) {
    int j = jj * 16 + jg;
    outb[(long)j * 512 + ii] = T[ii * 513 + j];
  }
}

// ====== per (b,h): Gram(k,q) over n=4096, l2-norm scaling, softmax, then
//        Wcomb[b, h*64+e, c] = sum_d S[d,e] * proj_w[h*64+d, c]            ======
__global__ __launch_bounds__(256) void head_attn_k(const float* __restrict__ Q,
                                                   const float* __restrict__ K,
                                                   const float* __restrict__ rescale,
                                                   const float* __restrict__ PW,
                                                   float* __restrict__ Wcomb) {
  __shared__ float kh[32 * 80];
  __shared__ float qh[32 * 80];
  __shared__ float G[64 * 65];
  __shared__ float red[512];          // [0:256) k-sumsq partials, [256:512) q
  __shared__ float nk[64], nq[64];
  int tid = threadIdx.x, lane = tid & 31, wave = tid >> 5;
  int blk = blockIdx.x, b = blk >> 3, h = blk & 7;
  const float* qb = Q + (long)b * 4096 * 512 + h * 64;
  const float* kb_ = K + (long)b * 4096 * 512 + h * 64;
  int wm = wave >> 1, wn = wave & 1;  // waves 4 (d) x 2 (e)
  const v8f z8 = {0.f, 0.f, 0.f, 0.f, 0.f, 0.f, 0.f, 0.f};
  v8f acc0 = z8, acc1 = z8;
  float pk = 0.f, pq = 0.f;
  int col = tid & 63, qgrp = tid >> 6;
  const int ml = lane & 15, kq = (lane >> 4) << 1;
  const int d0 = wm * 16, e0 = wn * 32;

  for (int kbi = 0; kbi < 128; kbi++) {
#pragma unroll
    for (int i = 0; i < 2; i++) {
      int f = tid + i * 256;
      int r = f >> 4, c4 = (f & 15) << 2;
      float4 vk = *(const float4*)(kb_ + (long)(kbi * 32 + r) * 512 + c4);
      float* pk_ = &kh[r * 80 + c4];
      pk_[0] = vk.x; pk_[1] = vk.y; pk_[2] = vk.z; pk_[3] = vk.w;
      float4 vq = *(const float4*)(qb + (long)(kbi * 32 + r) * 512 + c4);
      float* pq_ = &qh[r * 80 + c4];
      pq_[0] = vq.x; pq_[1] = vq.y; pq_[2] = vq.z; pq_[3] = vq.w;
    }
    __syncthreads();
#pragma unroll
    for (int r = 0; r < 8; r++) {
      float a = kh[(qgrp * 8 + r) * 80 + col]; pk += a * a;
      float c2 = qh[(qgrp * 8 + r) * 80 + col]; pq += c2 * c2;
    }
#pragma unroll
    for (int kk = 0; kk < 32; kk += 4) {
      v2f af, bf0, bf1;
      af.x = kh[(kk + kq) * 80 + d0 + ml];      af.y = kh[(kk + kq + 1) * 80 + d0 + ml];
      bf0.x = qh[(kk + kq) * 80 + e0 + ml];     bf0.y = qh[(kk + kq + 1) * 80 + e0 + ml];
      bf1.x = qh[(kk + kq) * 80 + e0 + 16 + ml];bf1.y = qh[(kk + kq + 1) * 80 + e0 + 16 + ml];
      acc0 = WMMA_F32(af, bf0, acc0);
      acc1 = WMMA_F32(af, bf1, acc1);
    }
    __syncthreads();
  }

  red[qgrp * 64 + col] = pk;
  red[256 + qgrp * 64 + col] = pq;
  const int mh = (lane >> 4) << 3, nl = lane & 15;
#pragma unroll
  for (int r = 0; r < 8; r++) {
    int d = d0 * 0 + wm * 16 + mh + r;  // wm*16 tile; mh in {0,8}
    // d range per wave: wm*16 .. wm*16+15
    G[d * 65 + e0 + nl] = acc0[r];
    G[d * 65 + e0 + 16 + nl] = acc1[r];
  }
  __syncthreads();
  if (tid < 64) {
    float s = red[tid] + red[64 + tid] + red[128 + tid] + red[192 + tid];
    nk[tid] = fmaxf(sqrtf(s), 1e-12f);
  } else if (tid < 128) {
    int c = tid - 64;
    float s = red[256 + c] + red[320 + c] + red[384 + c] + red[448 + c];
    nq[c] = fmaxf(sqrtf(s), 1e-12f);
  }
  __syncthreads();
  float rs = rescale[h];
  for (int rr = 0; rr < 8; rr++) {
    int d = wave * 8 + rr;
    float inkd = rs / nk[d];
    float v0 = G[d * 65 + lane] * inkd / nq[lane];
    float v1 = G[d * 65 + 32 + lane] * inkd / nq[32 + lane];
    float m = fmaxf(v0, v1);
#pragma unroll
    for (int o = 16; o >= 1; o >>= 1) m = fmaxf(m, __shfl_xor(m, o, 32));
    v0 = expf(v0 - m); v1 = expf(v1 - m);
    float sm = v0 + v1;
#pragma unroll
    for (int o = 16; o >= 1; o >>= 1) sm += __shfl_xor(sm, o, 32);
    float inv = 1.0f / sm;
    G[d * 65 + lane] = v0 * inv;
    G[d * 65 + 32 + lane] = v1 * inv;
  }
  __syncthreads();
  float* wc = Wcomb + (long)b * 262144 + (long)h * 64 * 512;
  for (int idx = tid; idx < 64 * 512; idx += 256) {
    int e = idx >> 9, c = idx & 511;
    const float* pwb = PW + (long)h * 64 * 512 + c;
    float a = 0.f;
#pragma unroll
    for (int d = 0; d < 64; d++) a += G[d * 65 + e] * pwb[d * 512];
    wc[(long)e * 512 + c] = a;
  }
}

// ================= depthwise 3x3 conv (NHWC, SAME), optional exact GELU ====
template <bool GELU>
__global__ __launch_bounds__(256) void dwconv3_k(const float* __restrict__ X,
                                                 const float* __restrict__ Wc,
                                                 float* __restrict__ Y) {
  long idx = (long)blockIdx.x * 256 + threadIdx.x;
  int c = (int)(idx & 511);
  long t = idx >> 9;
  int x = (int)(t & 63); t >>= 6;
  int y = (int)(t & 63);
  long bb = t >> 6;
  const float* base = X + bb * 4096 * 512;
  float acc = 0.f;
#pragma unroll
  for (int dy = -1; dy <= 1; dy++) {
    int yy = y + dy;
    if ((unsigned)yy < 64u) {
#pragma unroll
      for (int dx = -1; dx <= 1; dx++) {
        int xx = x + dx;
        if ((unsigned)xx < 64u)
          acc += base[((long)(yy * 64 + xx)) * 512 + c] * Wc[c * 9 + (dy + 1) * 3 + (dx + 1)];
      }
    }
  }
  if (GELU) acc = 0.5f * acc * (1.0f + erff(acc * 0.70710678118f));
  Y[idx] = acc;
}

// ============================== launcher ===================================
extern "C" void kernel_launch(void* const* d_in, const int* in_sizes, int n_in,
                              void* d_out, int out_size, void* d_ws, size_t ws_size,
                              hipStream_t stream) {
  (void)in_sizes; (void)n_in; (void)out_size; (void)ws_size;
  const float* x_in   = (const float*)d_in[0];
  const float* illu   = (const float*)d_in[1];
  const float* sem    = (const float*)d_in[2];
  const float* Wk_sif = (const float*)d_in[3];
  const float* Wq_sif = (const float*)d_in[4];
  const float* Wv_sif = (const float*)d_in[5];
  const float* ln_g   = (const float*)d_in[6];
  const float* ln_b   = (const float*)d_in[7];
  const float* Wq     = (const float*)d_in[8];
  const float* Wk     = (const float*)d_in[9];
  const float* Wv     = (const float*)d_in[10];
  const float* rescale= (const float*)d_in[11];
  const float* proj_w = (const float*)d_in[12];
  const float* proj_b = (const float*)d_in[13];
  const float* dw1    = (const float*)d_in[14];
  const float* dw2    = (const float*)d_in[15];
  float* out = (float*)d_out;

  char* ws = (char*)d_ws;
  const size_t BIG = 67108864ull;                 // 32768*512*4
  float* S0 = (float*)ws;                         // KnL -> Vs -> p1 -> q_inp
  float* S1 = (float*)(ws + BIG);                 // QnL -> v_inp -> k_inp
  float* S2 = (float*)(ws + 2 * BIG);             // vg
  float* W0 = (float*)(ws + 3 * BIG);             // Ab -> Wcomb (8 MB)
  float* W1 = (float*)(ws + 3 * BIG + 8388608ull);// SmT         (8 MB)

  dim3 blk(256);
  dim3 g1(256, 4, 1);   // unbatched GEMM: M=32768
  dim3 gz(32, 4, 8);    // batched GEMM: M=4096 per batch, per-batch W

  // --- SIF prior ---
  gemm512_wmma<0><<<g1, blk, 0, stream>>>(illu, Wk_sif, S0, 32768, 0, nullptr, nullptr, nullptr);
  gemm512_wmma<0><<<g1, blk, 0, stream>>>(sem,  Wq_sif, S1, 32768, 0, nullptr, nullptr, nullptr);
  ln_rows_k<<<4096, blk, 0, stream>>>(S0, ln_g, ln_b);                    // KnL
  ln_rows_k<<<4096, blk, 0, stream>>>(S1, ln_g, ln_b);                    // QnL
  sif_gram_wmma<<<dim3(4, 4, 8), blk, 0, stream>>>(S1, S0, W0);           // Ab
  softmaxT_k<<<dim3(32, 8), blk, 0, stream>>>(W0, W1);                    // SmT
  gemm512_wmma<0><<<g1, blk, 0, stream>>>(illu, Wv_sif, S0, 32768, 0, nullptr, nullptr, nullptr); // Vs
  gemm512_wmma<0><<<g1, blk, 0, stream>>>(x_in, Wv,     S1, 32768, 0, nullptr, nullptr, nullptr); // v_inp
  // vg = (Vs @ SmT[b] + Vs) * v_inp
  gemm512_wmma<1><<<gz, blk, 0, stream>>>(S0, W1, S2, 4096, 262144, S0, S1, nullptr);

  // --- positional conv branch (writes out_p into d_out) ---
  dwconv3_k<true ><<<65536, blk, 0, stream>>>(S1, dw1, S0);               // p1 = gelu(conv(v_inp))
  dwconv3_k<false><<<65536, blk, 0, stream>>>(S0, dw2, out);              // d_out = conv(p1)

  // --- attention ---
  gemm512_wmma<0><<<g1, blk, 0, stream>>>(x_in, Wq, S0, 32768, 0, nullptr, nullptr, nullptr); // q_inp
  gemm512_wmma<0><<<g1, blk, 0, stream>>>(x_in, Wk, S1, 32768, 0, nullptr, nullptr, nullptr); // k_inp
  head_attn_k<<<64, blk, 0, stream>>>(S0, S1, rescale, proj_w, W0);       // Wcomb
  // d_out = vg @ Wcomb[b] + proj_b + d_out
  gemm512_wmma<2><<<gz, blk, 0, stream>>>(S2, W0, out, 4096, 262144, nullptr, nullptr, proj_b);
}